// GaussianSplatting2D_47425028882642
// MI455X (gfx1250) — compile-verified
//
#include <hip/hip_runtime.h>
#include <math.h>

typedef __attribute__((ext_vector_type(2))) float v2f;
typedef __attribute__((ext_vector_type(8))) float v8f;

#define GS_N   1024
#define GS_W   256
#define GS_H   256
#define GS_EPS 0.3f

// -------------------------------------------------------------------------
// Kernel 1: per-Gaussian preprocessing.
// Produces, per Gaussian, the rank-6 bilinear-form coefficients of
//   power(p) = c0 + c1*gx + c2*gy + c3*gx^2 + c4*gy^2 + c5*gx*gy
// plus sigmoid(opacity), sigmoid(rgb), and camera-space depth.
// -------------------------------------------------------------------------
__global__ void gs_preprocess(const float* __restrict__ means,
                              const float* __restrict__ quats,
                              const float* __restrict__ scales,
                              const float* __restrict__ opac,
                              const float* __restrict__ rgbs,
                              const float* __restrict__ vm,
                              const float* __restrict__ Km,
                              float* __restrict__ g8,
                              float* __restrict__ zbuf) {
    int i = blockIdx.x * blockDim.x + threadIdx.x;
    if (i >= GS_N) return;

    float Rv[3][3] = {{vm[0], vm[1], vm[2]},
                      {vm[4], vm[5], vm[6]},
                      {vm[8], vm[9], vm[10]}};
    float t0 = vm[3], t1 = vm[7], t2 = vm[11];
    float fx = Km[0], cx = Km[2], fy = Km[4], cy = Km[5];

    float mx = means[3*i], my = means[3*i+1], mz = means[3*i+2];
    float px = Rv[0][0]*mx + Rv[0][1]*my + Rv[0][2]*mz + t0;
    float py = Rv[1][0]*mx + Rv[1][1]*my + Rv[1][2]*mz + t1;
    float pz = Rv[2][0]*mx + Rv[2][1]*my + Rv[2][2]*mz + t2;
    float invz = 1.0f / pz;
    float u = fx * px * invz + cx;
    float v = fy * py * invz + cy;

    // normalized quaternion -> rotation
    float qw = quats[4*i], qx = quats[4*i+1], qy = quats[4*i+2], qz = quats[4*i+3];
    float inq = 1.0f / (sqrtf(qw*qw + qx*qx + qy*qy + qz*qz) + 1e-8f);
    qw *= inq; qx *= inq; qy *= inq; qz *= inq;
    float Q[3][3];
    Q[0][0] = 1.f - 2.f*(qy*qy + qz*qz); Q[0][1] = 2.f*(qx*qy - qw*qz); Q[0][2] = 2.f*(qx*qz + qw*qy);
    Q[1][0] = 2.f*(qx*qy + qw*qz); Q[1][1] = 1.f - 2.f*(qx*qx + qz*qz); Q[1][2] = 2.f*(qy*qz - qw*qx);
    Q[2][0] = 2.f*(qx*qz - qw*qy); Q[2][1] = 2.f*(qy*qz + qw*qx); Q[2][2] = 1.f - 2.f*(qx*qx + qy*qy);

    float sc[3] = {scales[3*i], scales[3*i+1], scales[3*i+2]};
    float M[3][3];
    #pragma unroll
    for (int r = 0; r < 3; ++r)
        #pragma unroll
        for (int c = 0; c < 3; ++c)
            M[r][c] = Q[r][c] * sc[c];

    float S3[3][3];
    #pragma unroll
    for (int r = 0; r < 3; ++r)
        #pragma unroll
        for (int c = 0; c < 3; ++c)
            S3[r][c] = M[r][0]*M[c][0] + M[r][1]*M[c][1] + M[r][2]*M[c][2];

    // Sc = Rv * S3 * Rv^T
    float Tm[3][3];
    #pragma unroll
    for (int r = 0; r < 3; ++r)
        #pragma unroll
        for (int c = 0; c < 3; ++c)
            Tm[r][c] = Rv[r][0]*S3[0][c] + Rv[r][1]*S3[1][c] + Rv[r][2]*S3[2][c];
    float Sc[3][3];
    #pragma unroll
    for (int r = 0; r < 3; ++r)
        #pragma unroll
        for (int c = 0; c < 3; ++c)
            Sc[r][c] = Tm[r][0]*Rv[c][0] + Tm[r][1]*Rv[c][1] + Tm[r][2]*Rv[c][2];

    // J * Sc * J^T + eps*I
    float J00 = fx * invz, J02 = -fx * px * invz * invz;
    float J11 = fy * invz, J12 = -fy * py * invz * invz;
    float w00 = J00*Sc[0][0] + J02*Sc[2][0];
    float w01 = J00*Sc[0][1] + J02*Sc[2][1];
    float w02 = J00*Sc[0][2] + J02*Sc[2][2];
    float w11 = J11*Sc[1][1] + J12*Sc[2][1];
    float w12 = J11*Sc[1][2] + J12*Sc[2][2];
    float a = w00*J00 + w02*J02 + GS_EPS;
    float b = w01*J11 + w02*J12;
    float c = w11*J11 + w12*J12 + GS_EPS;

    float idet = 1.0f / (a*c - b*b);
    float ia = c * idet, ib = -b * idet, ic = a * idet;

    float op  = 1.0f / (1.0f + __expf(-opac[i]));
    float col = 1.0f / (1.0f + __expf(-rgbs[i]));

    float* o = g8 + 8*i;
    o[0] = -0.5f*(ia*u*u + ic*v*v) - ib*u*v;  // c0
    o[1] = ia*u + ib*v;                        // c1
    o[2] = ic*v + ib*u;                        // c2
    o[3] = -0.5f*ia;                           // c3
    o[4] = -0.5f*ic;                           // c4
    o[5] = -ib;                                // c5
    o[6] = op;
    o[7] = col;
    zbuf[i] = pz;
}

// -------------------------------------------------------------------------
// Kernel 2: stable depth sort (bitonic, index tie-break == stable argsort)
// plus gather of the 8-float structs into depth order.
// -------------------------------------------------------------------------
__global__ void gs_sort_gather(const float* __restrict__ zbuf,
                               const float* __restrict__ g8,
                               float* __restrict__ sg) {
    __shared__ float key[GS_N];
    __shared__ int   idx[GS_N];
    int t = threadIdx.x;
    key[t] = zbuf[t];
    idx[t] = t;
    __syncthreads();
    for (int k = 2; k <= GS_N; k <<= 1) {
        for (int j = k >> 1; j > 0; j >>= 1) {
            int ixj = t ^ j;
            if (ixj > t) {
                bool ascend = ((t & k) == 0);
                float ka = key[t], kb = key[ixj];
                int ia_ = idx[t], ib_ = idx[ixj];
                bool gt = (ka > kb) || (ka == kb && ia_ > ib_);
                if (gt == ascend) {
                    key[t] = kb; key[ixj] = ka;
                    idx[t] = ib_; idx[ixj] = ia_;
                }
            }
            __syncthreads();
        }
    }
    int src = idx[t];
    #pragma unroll
    for (int c = 0; c < 8; ++c)
        sg[t*8 + c] = g8[src*8 + c];
}

// -------------------------------------------------------------------------
// Kernel 3: render. Each wave owns 16 pixels; each 16-Gaussian chunk is
// evaluated as a rank-8 f32 GEMM tile via two chained V_WMMA_F32_16X16X4_F32
// (A = Gaussian coeffs [M=16 x K=8], B = pixel features [K=8 x N=16]).
// D layout: lanes 0-15 -> pixels x Gaussians 0-7 (VGPR rows), lanes 16-31 ->
// same pixels x Gaussians 8-15. Halves composite 8 Gaussians serially each,
// then are prefix-combined with shfl_xor(16).
// -------------------------------------------------------------------------
__global__ __launch_bounds__(256) void gs_render(const float* __restrict__ sg,
                                                 float* __restrict__ out) {
    __shared__ float gb[GS_N * 8];   // 32 KB: all sorted Gaussians resident
    int t = threadIdx.x;
    {
        const float4* s4 = (const float4*)sg;
        float4* d4 = (float4*)gb;
        for (int i = t; i < GS_N * 2; i += 256) d4[i] = s4[i];
    }
    __syncthreads();

    int lane = t & 31;
    int wave = t >> 5;
    int half = lane >> 4;          // 0: Gaussians M=0-7, 1: M=8-15
    int ln   = lane & 15;          // pixel column within tile (N index)
    int pid  = blockIdx.x * 128 + wave * 16 + ln;
    float gx = (float)(pid & (GS_W - 1)) + 0.5f;
    float gy = (float)(pid >> 8) + 0.5f;

    // B fragments: features [1, gx, gy, gx^2, gy^2, gx*gy, 0, 0]
    v2f b1, b2;
    b1.x = half ? gy      : 1.0f;
    b1.y = half ? gx * gx : gx;
    b2.x = half ? 0.0f    : gy * gy;
    b2.y = half ? 0.0f    : gx * gy;

    float T = 1.0f, img = 0.0f;

    for (int base = 0; base < GS_N; base += 16) {
        const float* s = &gb[(base + ln) * 8];
        v2f a1, a2;
        a1.x = half ? s[2] : s[0];
        a1.y = half ? s[3] : s[1];
        a2.x = half ? 0.0f : s[4];
        a2.y = half ? 0.0f : s[5];

        v8f d = {};
        d = __builtin_amdgcn_wmma_f32_16x16x4_f32(false, a1, false, b1,
                                                  (short)0, d, false, false);
        d = __builtin_amdgcn_wmma_f32_16x16x4_f32(false, a2, false, b2,
                                                  (short)0, d, false, false);

        // serial compositing of this lane's 8 Gaussians (in depth order)
        float S = 0.0f, P = 1.0f;
        int g0 = base + half * 8;
        #pragma unroll
        for (int r = 0; r < 8; ++r) {
            float power = d[r];
            float op  = gb[(g0 + r) * 8 + 6];
            float col = gb[(g0 + r) * 8 + 7];
            float al = op * __expf(power);
            al = fminf(al, 0.999f);
            bool keep = (power <= 0.0f) && (al >= (1.0f / 255.0f));
            al = keep ? al : 0.0f;
            S += P * al * col;
            P *= (1.0f - al);
        }

        // combine the two half-wave partials in depth order
        float So = __shfl_xor(S, 16, 32);
        float Po = __shfl_xor(P, 16, 32);
        float Slo = half ? So : S, Shi = half ? S : So;
        float Plo = half ? Po : P, Phi = half ? P : Po;
        img += T * (Slo + Plo * Shi);
        T   *= Plo * Phi;

        if (__all(T < 1e-6f)) break;   // wave-uniform early exit
    }

    if (half == 0 && pid < GS_W * GS_H) out[pid] = img;
}

// -------------------------------------------------------------------------
extern "C" void kernel_launch(void* const* d_in, const int* in_sizes, int n_in,
                              void* d_out, int out_size, void* d_ws, size_t ws_size,
                              hipStream_t stream) {
    const float* means = (const float*)d_in[0];
    const float* quats = (const float*)d_in[1];
    const float* scales = (const float*)d_in[2];
    const float* opac = (const float*)d_in[3];
    const float* rgbs = (const float*)d_in[4];
    const float* vm = (const float*)d_in[5];
    const float* Km = (const float*)d_in[6];
    float* outp = (float*)d_out;

    float* zbuf = (float*)d_ws;          // 1024 floats
    float* g8   = zbuf + GS_N;           // 8192 floats (unsorted structs)
    float* sg   = g8 + GS_N * 8;         // 8192 floats (depth-sorted structs)

    gs_preprocess<<<(GS_N + 255) / 256, 256, 0, stream>>>(
        means, quats, scales, opac, rgbs, vm, Km, g8, zbuf);
    gs_sort_gather<<<1, GS_N, 0, stream>>>(zbuf, g8, sg);
    gs_render<<<(GS_W * GS_H) / 128, 256, 0, stream>>>(sg, outp);
}